// AnchorPlusContrastiveLoss_10471130267821
// MI455X (gfx1250) — compile-verified
//
#include <hip/hip_runtime.h>
#include <hip/hip_bf16.h>
#include <math.h>

typedef __attribute__((ext_vector_type(2))) float v2f;
typedef __attribute__((ext_vector_type(8))) float v8f;

#define Bdim 4
#define Ndim 2048
#define Cdim 64
#define Kdim 32
#define BK   (Bdim * Kdim)        // 128 logit columns
#define ROWS (Bdim * Ndim)        // 8192 logit rows
#define INV_TEMP 0.1f
#define CE_W 10.0f

// ---- workspace layout (bytes) ----
#define ACC_OFF   0                                  // 3 doubles: nl_sum, mask_sum, ce_sum
#define P_OFF     256                                // ROWS*2 floats (emb + abs combined)
#define CN_OFF    (P_OFF + ROWS * 2 * 4)             // ROWS*Cdim floats (normalized contr_emb)
#define MEANS_OFF (CN_OFF + (size_t)ROWS * Cdim * 4) // BK*Cdim floats

// -------- init: zero accumulators, precompute p = embedding + abs_coords --------
__global__ void k_init(const float* __restrict__ emb, const float* __restrict__ abs_c,
                       double* __restrict__ acc, float* __restrict__ p) {
    int idx = blockIdx.x * blockDim.x + threadIdx.x;
    if (idx < 3) acc[idx] = 0.0;
    if (idx < ROWS) {
        p[idx * 2 + 0] = emb[idx * 2 + 0] + abs_c[idx * 2 + 0];
        p[idx * 2 + 1] = emb[idx * 2 + 1] + abs_c[idx * 2 + 1];
    }
}

// -------- anchor term: one block per (b, i) row, coalesced mask reads --------
__global__ void k_anchor(const float* __restrict__ p, const int* __restrict__ mask,
                         double* __restrict__ acc) {
    int bi = blockIdx.x;               // b*N + i
    int b  = bi / Ndim;
    float pix = p[bi * 2 + 0];
    float piy = p[bi * 2 + 1];
    const int*   mrow = mask + (size_t)bi * Ndim;
    const float* pb   = p + (size_t)b * Ndim * 2;

    float nl = 0.0f, ct = 0.0f;
    for (int j = threadIdx.x; j < Ndim; j += blockDim.x) {
        if (mrow[j] == 1) {
            float dx = pix - pb[j * 2 + 0];
            float dy = piy - pb[j * 2 + 1];
            float d2 = dx * dx + dy * dy;
            nl += 1.0f - __expf(-d2 * INV_TEMP);
            ct += 1.0f;
        }
    }
    for (int s = 16; s >= 1; s >>= 1) { nl += __shfl_xor(nl, s); ct += __shfl_xor(ct, s); }
    __shared__ float snl[8], sct[8];
    int wave = threadIdx.x >> 5, lane = threadIdx.x & 31;
    if (lane == 0) { snl[wave] = nl; sct[wave] = ct; }
    __syncthreads();
    if (wave == 0) {
        nl = (lane < 8) ? snl[lane] : 0.0f;
        ct = (lane < 8) ? sct[lane] : 0.0f;
        for (int s = 4; s >= 1; s >>= 1) { nl += __shfl_xor(nl, s); ct += __shfl_xor(ct, s); }
        if (lane == 0) { atomicAdd(&acc[0], (double)nl); atomicAdd(&acc[1], (double)ct); }
    }
}

// -------- L2-normalize contr_emb rows: one wave per row --------
__global__ void k_norm(const float* __restrict__ ce, float* __restrict__ cn) {
    int wave = threadIdx.x >> 5, lane = threadIdx.x & 31;
    int row = blockIdx.x * (blockDim.x >> 5) + wave;
    if (row >= ROWS) return;
    const float* x = ce + (size_t)row * Cdim;
    float x0 = x[lane], x1 = x[lane + 32];
    float ss = x0 * x0 + x1 * x1;
    for (int s = 16; s >= 1; s >>= 1) ss += __shfl_xor(ss, s);
    float inv = 1.0f / fmaxf(sqrtf(ss), 1e-12f);
    float* o = cn + (size_t)row * Cdim;
    o[lane] = x0 * inv;
    o[lane + 32] = x1 * inv;
}

// -------- per-batch cluster means: one block per (b,k), one thread per channel --------
__global__ void k_means(const float* __restrict__ cn, const int* __restrict__ labels,
                        float* __restrict__ means) {
    int bk = blockIdx.x;
    int b = bk / Kdim, k = bk % Kdim;
    int c = threadIdx.x;   // 64 threads
    const int*   lab  = labels + b * Ndim;
    const float* base = cn + (size_t)b * Ndim * Cdim;
    float s = 0.0f, cnt = 0.0f;
    for (int n = 0; n < Ndim; ++n) {
        if (lab[n] == k) { s += base[(size_t)n * Cdim + c]; cnt += 1.0f; }
    }
    means[bk * Cdim + c] = s / fmaxf(cnt, 1.0f);
}

// -------- logits GEMM via V_WMMA_F32_16X16X4_F32 + fused log-softmax NLL --------
// block = 128 threads (4 waves); block covers 64 rows, wave covers 16 rows.
__global__ void __launch_bounds__(128) k_ce(const float* __restrict__ cn,
                                            const float* __restrict__ means,
                                            const int* __restrict__ labels,
                                            double* __restrict__ acc) {
    __shared__ float sB[Cdim * BK];   // B matrix, [c][col], 32 KB
    __shared__ float sL[64 * BK];     // logits for this block's 64 rows, 32 KB

    int tid = threadIdx.x;
    // stage B = means^T into LDS: sB[c*128 + col] = means[col*64 + c]
    for (int idx = tid; idx < Cdim * BK; idx += 128) {
        int col = idx >> 6, cc = idx & 63;
        sB[cc * BK + col] = means[col * Cdim + cc];
    }
    __syncthreads();

    int wave = tid >> 5, lane = tid & 31;
    int m = lane & 15;                 // row within 16x16 tile / col index for B frag
    int khalf = lane >> 4;             // 0: K={kb,kb+1}, 1: K={kb+2,kb+3}
    int row0 = blockIdx.x * 64 + wave * 16;

    // A fragments for all 16 K-steps (K=64, step 4): reused across 8 column tiles
    const float* Ar = cn + (size_t)(row0 + m) * Cdim;
    v2f a[16];
#pragma unroll
    for (int kk = 0; kk < 16; ++kk) {
        int kb = kk * 4 + khalf * 2;
        a[kk].x = Ar[kb];
        a[kk].y = Ar[kb + 1];
    }

#pragma unroll
    for (int ctile = 0; ctile < 8; ++ctile) {
        int col = ctile * 16 + m;
        v8f accv = {};
#pragma unroll
        for (int kk = 0; kk < 16; ++kk) {
            int kb = kk * 4 + khalf * 2;
            v2f bb;
            bb.x = sB[kb * BK + col];
            bb.y = sB[(kb + 1) * BK + col];
            accv = __builtin_amdgcn_wmma_f32_16x16x4_f32(
                false, a[kk], false, bb, (short)0, accv, false, false);
        }
        // D layout: VGPR v -> row v (lanes 0-15) / row v+8 (lanes 16-31), col = lane&15
#pragma unroll
        for (int v = 0; v < 8; ++v) {
            int r = v + khalf * 8;
            sL[(wave * 16 + r) * BK + col] = accv[v];
        }
    }
    // DS ops from the same wave are in-order: the wave reads back only its own strip.

    // log-softmax + NLL for this wave's 16 rows (128 columns, 4 per lane)
    float wsum = 0.0f;
    for (int r = 0; r < 16; ++r) {
        const float* rowp = sL + (wave * 16 + r) * BK;
        float v0 = rowp[lane], v1 = rowp[lane + 32], v2 = rowp[lane + 64], v3 = rowp[lane + 96];
        float mx = fmaxf(fmaxf(v0, v1), fmaxf(v2, v3));
        for (int s = 16; s >= 1; s >>= 1) mx = fmaxf(mx, __shfl_xor(mx, s));
        float se = __expf(v0 - mx) + __expf(v1 - mx) + __expf(v2 - mx) + __expf(v3 - mx);
        for (int s = 16; s >= 1; s >>= 1) se += __shfl_xor(se, s);
        int t = labels[row0 + r];          // targets in [0, 32): no batch offset, per reference
        float xt = rowp[t];
        wsum += mx + __logf(se) - xt;      // -log_softmax at target
    }
    if (lane == 0) atomicAdd(&acc[2], (double)wsum);
}

// -------- finalize: scalar combine --------
__global__ void k_final(const double* __restrict__ acc, float* __restrict__ out) {
    if (threadIdx.x == 0 && blockIdx.x == 0) {
        double anchor = acc[0] / acc[1];
        double ce = acc[2] / (double)ROWS;
        out[0] = (float)(anchor + (double)CE_W * ce);
    }
}

extern "C" void kernel_launch(void* const* d_in, const int* in_sizes, int n_in,
                              void* d_out, int out_size, void* d_ws, size_t ws_size,
                              hipStream_t stream) {
    const float* emb    = (const float*)d_in[0];
    const float* ce     = (const float*)d_in[1];
    const float* abs_c  = (const float*)d_in[2];
    const int*   mask   = (const int*)d_in[3];
    const int*   labels = (const int*)d_in[4];
    float* out = (float*)d_out;

    char* w = (char*)d_ws;
    double* acc  = (double*)(w + ACC_OFF);
    float* p     = (float*)(w + P_OFF);
    float* cn    = (float*)(w + CN_OFF);
    float* means = (float*)(w + MEANS_OFF);

    k_init  <<<(ROWS + 255) / 256, 256, 0, stream>>>(emb, abs_c, acc, p);
    k_anchor<<<ROWS,               256, 0, stream>>>(p, mask, acc);
    k_norm  <<<ROWS / 8,           256, 0, stream>>>(ce, cn);
    k_means <<<BK,                  64, 0, stream>>>(cn, labels, means);
    k_ce    <<<ROWS / 64,          128, 0, stream>>>(cn, means, labels, acc);
    k_final <<<1,                   32, 0, stream>>>(acc, out);
}